// SamplePatches_21706764714731
// MI455X (gfx1250) — compile-verified
//
#include <hip/hip_runtime.h>
#include <stdint.h>
#include <stddef.h>

#define BATCH   8
#define CH      3
#define HL      256
#define WL      256
#define HH      2048
#define WH      2048
#define NP      16
#define PSZ     128
#define AN      (HL * WL)          // 65536 logits per batch
#define PATCH_ELEMS ((size_t)BATCH * NP * CH * PSZ * PSZ)   // 6291456

typedef uint32_t u32x4 __attribute__((ext_vector_type(4)));
typedef uint32_t u32x8 __attribute__((ext_vector_type(8)));

// ---------------------------------------------------------------------------
// helpers
// ---------------------------------------------------------------------------
__device__ __forceinline__ float gumbel_key(float a, float u) {
    // logits + gumbel noise:  log(a) + (-log(-log(u)))
    return logf(a) - logf(-logf(u));
}

__device__ __forceinline__ void wait_asynccnt0() {
#if __has_builtin(__builtin_amdgcn_s_wait_asynccnt)
    __builtin_amdgcn_s_wait_asynccnt(0);
#else
    asm volatile("s_wait_asynccnt 0" ::: "memory");
#endif
}

__device__ __forceinline__ void wait_tensorcnt0() {
#if __has_builtin(__builtin_amdgcn_s_wait_tensorcnt)
    __builtin_amdgcn_s_wait_tensorcnt(0);
#else
    asm volatile("s_wait_tensorcnt 0" ::: "memory");
#endif
}

__device__ __forceinline__ uint32_t rfl(uint32_t v) {
    return (uint32_t)__builtin_amdgcn_readfirstlane((int)v);
}

// Build a 2D TDM descriptor (D# groups 0 and 1, CDNA5 ISA).
// LDS side is a contiguous w*h tile; global side is a strided 2D walk.
// Tensor dims are set large: we pre-clip the tile, so no OOB reliance.
__device__ __forceinline__ void tdm_build_2d(uint32_t lds_addr, uint64_t gaddr,
                                             uint32_t tile_w, uint32_t tile_h,
                                             uint64_t row_stride_elems,
                                             u32x4* g0, u32x8* g1) {
    const uint32_t tw = 1u << 20;        // tensor_dim0 (huge, never OOB)
    const uint32_t th = 1u << 20;        // tensor_dim1
    (*g0)[0] = rfl(1u);                                          // count=1
    (*g0)[1] = rfl(lds_addr);                                    // lds_addr
    (*g0)[2] = rfl((uint32_t)gaddr);                             // gaddr[31:0]
    (*g0)[3] = rfl(((uint32_t)(gaddr >> 32) & 0x01ffffffu)       // gaddr[56:32]
                   | 0x80000000u);                               // type=2
    (*g1)[0] = rfl(2u << 16);                                    // data_size=4B
    (*g1)[1] = rfl((tw & 0xffffu) << 16);                        // dim0[15:0]
    (*g1)[2] = rfl((tw >> 16) | ((th & 0xffffu) << 16));         // dim0[31:16], dim1[15:0]
    (*g1)[3] = rfl((th >> 16) | (tile_w << 16));                 // dim1[31:16], tile_dim0
    (*g1)[4] = rfl(tile_h & 0xffffu);                            // tile_dim1 (tile_dim2=0)
    (*g1)[5] = rfl((uint32_t)row_stride_elems);                  // dim0_stride[31:0]
    (*g1)[6] = rfl((uint32_t)(row_stride_elems >> 32) & 0xffffu);// dim0_stride[47:32]
    (*g1)[7] = rfl(0u);
}

__device__ __forceinline__ void tdm_load_2d(uint32_t lds_addr, uint64_t gaddr,
                                            uint32_t w, uint32_t h, uint64_t stride) {
    u32x4 g0; u32x8 g1;
    tdm_build_2d(lds_addr, gaddr, w, h, stride, &g0, &g1);
    asm volatile("tensor_load_to_lds %0, %1" :: "s"(g0), "s"(g1) : "memory");
}

__device__ __forceinline__ void tdm_store_2d(uint32_t lds_addr, uint64_t gaddr,
                                             uint32_t w, uint32_t h, uint64_t stride) {
    u32x4 g0; u32x8 g1;
    tdm_build_2d(lds_addr, gaddr, w, h, stride, &g0, &g1);
    asm volatile("tensor_store_from_lds %0, %1" :: "s"(g0), "s"(g1) : "memory");
}

// ---------------------------------------------------------------------------
// kernel 1 (optional): precompute gumbel keys into workspace
// ---------------------------------------------------------------------------
__global__ void gumbel_keys_kernel(const float* __restrict__ att,
                                   const float* __restrict__ u,
                                   float* __restrict__ keys) {
    const int total = BATCH * AN;
    for (int i = blockIdx.x * blockDim.x + threadIdx.x; i < total;
         i += gridDim.x * blockDim.x) {
        keys[i] = gumbel_key(att[i], u[i]);
    }
}

// ---------------------------------------------------------------------------
// kernel 2: top-16 per batch via successive lexicographic max selection.
// (value descending, index ascending) == jax.lax.top_k ordering.
// ---------------------------------------------------------------------------
__global__ void __launch_bounds__(1024)
topk_kernel(const float* __restrict__ att,
            const float* __restrict__ u,
            const float* __restrict__ keys,     // may be nullptr
            int* __restrict__ offs,
            float* __restrict__ samp_att) {
    const int b   = blockIdx.x;
    const int tid = threadIdx.x;

    __shared__ float sval[1024];
    __shared__ int   sidx[1024];

    const float* attb = att + (size_t)b * AN;
    const float* ub   = u   + (size_t)b * AN;
    const float* kb   = keys ? keys + (size_t)b * AN : nullptr;

    float prevV = 3.4e38f;
    int   prevI = -1;

    for (int r = 0; r < NP; ++r) {
        float best = -3.4e38f;
        int   bi   = 0x7fffffff;
        for (int e = tid; e < AN; e += 1024) {
            const float k = kb ? kb[e] : gumbel_key(attb[e], ub[e]);
            const bool ok = (k < prevV) || (k == prevV && e > prevI);
            if (ok && (k > best || (k == best && e < bi))) { best = k; bi = e; }
        }
        sval[tid] = best; sidx[tid] = bi;
        __syncthreads();
        for (int s = 512; s > 0; s >>= 1) {
            if (tid < s) {
                const float ov = sval[tid + s]; const int oi = sidx[tid + s];
                const float mv = sval[tid];     const int mi = sidx[tid];
                if (ov > mv || (ov == mv && oi < mi)) { sval[tid] = ov; sidx[tid] = oi; }
            }
            __syncthreads();
        }
        const float selV = sval[0];
        const int   selI = sidx[0];
        if (tid == 0) {
            samp_att[b * NP + r] = attb[selI];
            const int row = selI >> 8;
            const int col = selI & 255;
            offs[(b * NP + r) * 2 + 0] = row * 8 - 60;   // exact: round(8s+4-64)
            offs[(b * NP + r) * 2 + 1] = col * 8 - 60;
        }
        prevV = selV; prevI = selI;
        __syncthreads();
    }
}

// ---------------------------------------------------------------------------
// kernel 3 (primary): patch extraction via the Tensor Data Mover.
// One block per (b, n, c). The in-bounds sub-rectangle of the patch is moved
// by two TDM descriptors (global->LDS strided gather, LDS->global store);
// threads only zero the out-of-image border (empty for interior patches).
// ---------------------------------------------------------------------------
__global__ void __launch_bounds__(128)
patch_extract_tdm(const float* __restrict__ x_high,
                  const int* __restrict__ offs,
                  float* __restrict__ out) {
    __shared__ float buf[PSZ * PSZ];         // 64 KB tile staging

    const int blk = blockIdx.x;              // b*48 + n*3 + c
    const int c   = blk % CH;
    const int n   = (blk / CH) % NP;
    const int b   = blk / (CH * NP);

    const int py0 = offs[(b * NP + n) * 2 + 0];
    const int px0 = offs[(b * NP + n) * 2 + 1];

    const float* __restrict__ src = x_high + ((size_t)(b * CH + c)) * HH * WH;
    float* __restrict__ dst = out + (size_t)blk * PSZ * PSZ;

    // clip patch rectangle to image (always non-empty: offsets in [-60,1980])
    const int ys = py0 < 0 ? -py0 : 0;
    const int xs = px0 < 0 ? -px0 : 0;
    const int ye = (py0 + PSZ > HH) ? (HH - py0) : PSZ;
    const int xe = (px0 + PSZ > WH) ? (WH - px0) : PSZ;
    const uint32_t w = (uint32_t)(xe - xs);
    const uint32_t h = (uint32_t)(ye - ys);

    // zero out-of-image border (no-op for interior patches)
    if (ys > 0 || xs > 0 || ye < PSZ || xe < PSZ) {
        for (int i = threadIdx.x; i < PSZ * PSZ; i += 128) {
            const int y = i >> 7;
            const int x = i & (PSZ - 1);
            if (y < ys || y >= ye || x < xs || x >= xe) dst[i] = 0.0f;
        }
    }

    // DMA the clipped tile: global (stride WH) -> LDS -> global (stride PSZ)
    if (threadIdx.x == 0) {
        const uint32_t lds_base = (uint32_t)(uintptr_t)(&buf[0]);
        const uint64_t gsrc = (uint64_t)(uintptr_t)
            (src + (size_t)(py0 + ys) * WH + (px0 + xs));
        tdm_load_2d(lds_base, gsrc, w, h, WH);
        wait_tensorcnt0();                   // load done before store reads LDS
        const uint64_t gdst = (uint64_t)(uintptr_t)
            (dst + (size_t)ys * PSZ + xs);
        tdm_store_2d(lds_base, gdst, w, h, PSZ);
        wait_tensorcnt0();
    }
}

// ---------------------------------------------------------------------------
// kernel 3b (kept compiled; async global<->LDS b128 path)
// ---------------------------------------------------------------------------
__global__ void __launch_bounds__(128)
patch_extract_async(const float* __restrict__ x_high,
                    const int* __restrict__ offs,
                    float* __restrict__ out) {
    const int blk = blockIdx.x;
    const int c   = blk % CH;
    const int n   = (blk / CH) % NP;
    const int b   = blk / (CH * NP);

    const int py0 = offs[(b * NP + n) * 2 + 0];
    const int px0 = offs[(b * NP + n) * 2 + 1];

    const float* __restrict__ src = x_high + ((size_t)(b * CH + c)) * HH * WH;
    float* __restrict__ dst = out + (size_t)blk * PSZ * PSZ;

    const int tid  = threadIdx.x;
    const bool interior = (py0 >= 0) && (py0 + PSZ <= HH) &&
                          (px0 >= 0) && (px0 + PSZ <= WH);

    __shared__ float buf[4][PSZ];
    const int w    = tid >> 5;
    const int lane = tid & 31;

    if (interior) {
        const uint32_t lds_addr = (uint32_t)(uintptr_t)(&buf[w][lane * 4]);
        for (int ybase = 0; ybase < PSZ; ybase += 4) {
            const int y = ybase + w;
            const uint64_t ga = (uint64_t)(uintptr_t)
                (src + (size_t)(py0 + y) * WH + px0 + lane * 4);
            const uint64_t oa = (uint64_t)(uintptr_t)
                (dst + (size_t)y * PSZ + lane * 4);
            asm volatile("global_load_async_to_lds_b128 %0, %1, off"
                         :: "v"(lds_addr), "v"(ga) : "memory");
            wait_asynccnt0();
            asm volatile("global_store_async_from_lds_b128 %0, %1, off"
                         :: "v"(oa), "v"(lds_addr) : "memory");
            wait_asynccnt0();
        }
    } else {
        for (int i = tid; i < PSZ * PSZ; i += 128) {
            const int y  = i >> 7;
            const int x  = i & (PSZ - 1);
            const int py = py0 + y;
            const int px = px0 + x;
            float v = 0.0f;
            if (py >= 0 && py < HH && px >= 0 && px < WH)
                v = src[(size_t)py * WH + px];
            dst[i] = v;
        }
    }
}

// ---------------------------------------------------------------------------
// launcher
// ---------------------------------------------------------------------------
extern "C" void kernel_launch(void* const* d_in, const int* in_sizes, int n_in,
                              void* d_out, int out_size, void* d_ws, size_t ws_size,
                              hipStream_t stream) {
    (void)in_sizes; (void)n_in; (void)out_size;

    // setup_inputs order: x_low, x_high, attention, noise_u
    const float* x_high    = (const float*)d_in[1];
    const float* attention = (const float*)d_in[2];
    const float* noise_u   = (const float*)d_in[3];
    float* out = (float*)d_out;

    // workspace: [0,1024) offsets (256 ints), then optional gumbel keys
    int* offs = (int*)d_ws;
    const size_t keys_bytes = (size_t)BATCH * AN * sizeof(float);
    float* keys = nullptr;
    if (ws_size >= 1024 + keys_bytes) {
        keys = (float*)((char*)d_ws + 1024);
        hipLaunchKernelGGL(gumbel_keys_kernel, dim3(512), dim3(256), 0, stream,
                           attention, noise_u, keys);
    }

    float* samp_att_out = out + PATCH_ELEMS;   // tail of d_out: [B, NP]
    hipLaunchKernelGGL(topk_kernel, dim3(BATCH), dim3(1024), 0, stream,
                       attention, noise_u, keys, offs, samp_att_out);

    hipLaunchKernelGGL(patch_extract_tdm, dim3(BATCH * NP * CH), dim3(128), 0,
                       stream, x_high, offs, out);
}